// HungarianMatcher_37099927503402
// MI455X (gfx1250) — compile-verified
//
#include <hip/hip_runtime.h>
#include <stdint.h>

#define B_ 16
#define Q_ 500
#define N_ 128
#define KD_ 34          // 17 keypoints * 2
#define EPS_WH 1e-4f

#define QCHUNK 50
#define QBLOCKS (Q_ / QCHUNK)   // 10 query-tiles per batch

// ---------------- CDNA5 async global->LDS helpers -------------------------

__device__ __forceinline__ void async_dword_to_lds(const float* g, float* l) {
#if defined(__gfx1250__)
  // LDS aperture: low 32 bits of a generic LDS pointer are the wave-relative
  // LDS byte address the async-DMA path expects in VDST.
  unsigned int lds_addr = (unsigned int)(uintptr_t)l;
  unsigned long long gaddr = (unsigned long long)(uintptr_t)g;
  asm volatile("global_load_async_to_lds_b32 %0, %1, off"
               :: "v"(lds_addr), "v"(gaddr)
               : "memory");
#else
  *l = *g;
#endif
}

__device__ __forceinline__ void wait_asynccnt0() {
#if defined(__gfx1250__)
#if __has_builtin(__builtin_amdgcn_s_wait_asynccnt)
  __builtin_amdgcn_s_wait_asynccnt(0);
#else
  asm volatile("s_wait_asynccnt 0" ::: "memory");
#endif
#endif
}

// ---------------- Phase 1: cost tensor ------------------------------------
// grid = B*QBLOCKS blocks of 256 threads (8 wave32).
// Threads = (sub in {0,1}) x (n in 0..127). Target data staged to LDS via
// async DMA once per block; stride-34 LDS rows are bank-conflict-free in
// wave32 (17 is invertible mod 32 -> 32 distinct banks).

__global__ void __launch_bounds__(256) hm_cost_kernel(
    const float* __restrict__ PB,   // (B,Q,4)  cxcywh
    const float* __restrict__ PK,   // (B,Q,34)
    const float* __restrict__ TB,   // (B,N,4)
    const float* __restrict__ TK,   // (B,N,34)
    float* __restrict__ C,          // (B,Q,N)
    float* __restrict__ CT,         // (B,N,Q) workspace copy (may be null)
    int writeCT)
{
  __shared__ float smem[N_ * 4 + N_ * KD_ + N_ * 5];
  float* s_tbox = smem;                     // 512 raw cxcywh
  float* s_tkpt = smem + N_ * 4;            // 4352
  float* s_txy  = smem + N_ * 4 + N_ * KD_; // 640: x0,y0,x1,y1,area

  const int b   = blockIdx.x / QBLOCKS;
  const int q0  = (blockIdx.x % QBLOCKS) * QCHUNK;
  const int tid = threadIdx.x;

  // Async-stage target boxes + keypoints for this batch image (4864 dwords).
  const int tot = N_ * 4 + N_ * KD_;
  for (int idx = tid; idx < tot; idx += 256) {
    const float* src = (idx < N_ * 4) ? (TB + b * (N_ * 4) + idx)
                                      : (TK + b * (N_ * KD_) + (idx - N_ * 4));
    async_dword_to_lds(src, &smem[idx]);
  }
  wait_asynccnt0();
  __syncthreads();

  // Precompute target xyxy + area (clamped boxes), one thread per target.
  if (tid < N_) {
    float cx = s_tbox[tid * 4 + 0], cy = s_tbox[tid * 4 + 1];
    float w  = s_tbox[tid * 4 + 2], h  = s_tbox[tid * 4 + 3];
    float x0 = cx - 0.5f * w, y0 = cy - 0.5f * h;
    float x1 = fmaxf(cx + 0.5f * w, x0 + EPS_WH);
    float y1 = fmaxf(cy + 0.5f * h, y0 + EPS_WH);
    s_txy[tid * 5 + 0] = x0;
    s_txy[tid * 5 + 1] = y0;
    s_txy[tid * 5 + 2] = x1;
    s_txy[tid * 5 + 3] = y1;
    s_txy[tid * 5 + 4] = (x1 - x0) * (y1 - y0);
  }
  __syncthreads();

  const int n   = tid & (N_ - 1);
  const int sub = tid >> 7;

  for (int qq = sub; qq < QCHUNK; qq += 2) {
    const int q = q0 + qq;

    const float* pb = PB + (size_t)(b * Q_ + q) * 4;
    float pcx = pb[0], pcy = pb[1], pw = pb[2], ph = pb[3];
    float px0 = pcx - 0.5f * pw, py0 = pcy - 0.5f * ph;
    float px1 = fmaxf(pcx + 0.5f * pw, px0 + EPS_WH);
    float py1 = fmaxf(pcy + 0.5f * ph, py0 + EPS_WH);
    float ap  = (px1 - px0) * (py1 - py0);

    const float* pk = PK + (size_t)(b * Q_ + q) * KD_;
    float pkr[KD_];
#pragma unroll
    for (int k = 0; k < KD_; ++k) pkr[k] = pk[k];

    // L1 bbox cost on raw cxcywh
    float cb = fabsf(pcx - s_tbox[n * 4 + 0]) + fabsf(pcy - s_tbox[n * 4 + 1]) +
               fabsf(pw  - s_tbox[n * 4 + 2]) + fabsf(ph  - s_tbox[n * 4 + 3]);

    // GIoU cost
    float tx0 = s_txy[n * 5 + 0], ty0 = s_txy[n * 5 + 1];
    float tx1 = s_txy[n * 5 + 2], ty1 = s_txy[n * 5 + 3];
    float at  = s_txy[n * 5 + 4];
    float wx    = fmaxf(fminf(px1, tx1) - fmaxf(px0, tx0), 0.0f);
    float wy    = fmaxf(fminf(py1, ty1) - fmaxf(py0, ty0), 0.0f);
    float inter = wx * wy;
    float uni   = ap + at - inter;
    float iou   = inter / uni;
    float ex    = fmaxf(fmaxf(px1, tx1) - fminf(px0, tx0), 0.0f);
    float ey    = fmaxf(fmaxf(py1, ty1) - fminf(py0, ty0), 0.0f);
    float enc   = ex * ey;
    float giou  = iou - (enc - uni) / enc;

    // keypoint L1 cost over 34 dims
    float ck = 0.0f;
#pragma unroll
    for (int k = 0; k < KD_; ++k) ck += fabsf(pkr[k] - s_tkpt[n * KD_ + k]);

    float cost = 5.0f * cb + 2.0f * (-giou) + ck;
    C[(size_t)(b * Q_ + q) * N_ + n] = cost;
    if (writeCT)
      CT[(size_t)(b * N_ + n) * Q_ + q] = cost;   // transposed copy for JV
  }
}

// ---------------- Phase 2: exact JV assignment per image ------------------
// One 512-thread block per batch image. Works on cost.T (rows = targets,
// cols = queries) in float64, matching the reference LSA control flow:
// strict-< relaxation, argmin tie -> smallest unscanned column, duals
// updated before augmentation. Argmin uses a wave32 shfl butterfly (no
// barriers) + a 16-partial cross-wave combine (2 barriers per iteration).

template <bool TRANSPOSED>
__global__ void __launch_bounds__(512) hm_hungarian_kernel(
    const float* __restrict__ CSRC, // TRANSPOSED ? (B,N,Q) : (B,Q,N)
    float* __restrict__ outR,       // (B,N) row indices (queries), as float
    float* __restrict__ outC)       // (B,N) col indices (targets), as float
{
  __shared__ double u[N_];
  __shared__ double v[Q_];
  __shared__ double spc[Q_];
  __shared__ int    path[Q_];
  __shared__ int    row4col[Q_];
  __shared__ int    col4row[N_];
  __shared__ int    SCm[Q_];
  __shared__ int    SRm[N_];
  __shared__ double redv[16];
  __shared__ int    redi[16];
  __shared__ double s_minval;
  __shared__ int    s_i, s_sink;

  const int b   = blockIdx.x;
  const int tid = threadIdx.x;
  const float* Cb = CSRC + (size_t)b * Q_ * N_;
  const double DINF = (double)__builtin_inf();

  // Pre-warm this image's 256KB cost slice into cache (global_prefetch).
  for (int off = tid * 16; off < Q_ * N_; off += 512 * 16)
    __builtin_prefetch(Cb + off, 0, 0);

  for (int j = tid; j < Q_; j += 512) { v[j] = 0.0; row4col[j] = -1; }
  if (tid < N_) { u[tid] = 0.0; col4row[tid] = -1; }
  __syncthreads();

  for (int cur = 0; cur < N_; ++cur) {
    for (int j = tid; j < Q_; j += 512) { path[j] = -1; SCm[j] = 0; spc[j] = DINF; }
    if (tid < N_) SRm[tid] = 0;
    if (tid == 0) { s_minval = 0.0; s_i = cur; s_sink = -1; }
    __syncthreads();

    while (true) {
      const int    i  = s_i;
      const double mv = s_minval;
      if (tid == 0) SRm[i] = 1;

      // relax this thread's column for row i (reads own spc only)
      double val = DINF;
      int    idx = tid;
      if (tid < Q_ && !SCm[tid]) {
        double c = TRANSPOSED ? (double)Cb[(size_t)i * Q_ + tid]     // contiguous
                              : (double)Cb[(size_t)tid * N_ + i];    // strided
        double r = ((mv + c) - u[i]) - v[tid];
        if (r < spc[tid]) { spc[tid] = r; path[tid] = i; }
        val = spc[tid];
      }

      // wave32 butterfly argmin (ties -> smallest column index)
#pragma unroll
      for (int off = 16; off > 0; off >>= 1) {
        double v2 = __shfl_xor(val, off, 32);
        int    i2 = __shfl_xor(idx, off, 32);
        if (v2 < val || (v2 == val && i2 < idx)) { val = v2; idx = i2; }
      }
      if ((tid & 31) == 0) { redv[tid >> 5] = val; redi[tid >> 5] = idx; }
      __syncthreads();

      if (tid < 32) {
        double v3 = (tid < 16) ? redv[tid] : DINF;
        int    i3 = (tid < 16) ? redi[tid] : 0x7fffffff;
#pragma unroll
        for (int off = 8; off > 0; off >>= 1) {
          double v2 = __shfl_xor(v3, off, 32);
          int    i2 = __shfl_xor(i3, off, 32);
          if (v2 < v3 || (v2 == v3 && i2 < i3)) { v3 = v2; i3 = i2; }
        }
        if (tid == 0) {
          s_minval = v3;
          SCm[i3] = 1;
          if (row4col[i3] < 0) s_sink = i3;
          else                 s_i = row4col[i3];
        }
      }
      __syncthreads();
      if (s_sink >= 0) break;
    }

    // dual update (before augmentation, as in the reference)
    const double mv = s_minval;
    if (tid == 0) u[cur] += mv;
    if (tid < N_ && tid != cur && SRm[tid]) u[tid] += mv - spc[col4row[tid]];
    for (int j = tid; j < Q_; j += 512)
      if (SCm[j]) v[j] -= mv - spc[j];
    __syncthreads();

    // augment along the alternating path (serial, <=128 steps)
    if (tid == 0) {
      int j = s_sink;
      while (true) {
        int i2 = path[j];
        row4col[j] = i2;
        int t = col4row[i2];
        col4row[i2] = j;
        j = t;
        if (i2 == cur) break;
      }
    }
    __syncthreads();
  }

  // col4row[t] = query assigned to target t (unique values) ->
  // rank-by-count reproduces np.argsort; emit (rows, cols) as floats.
  if (tid < N_) {
    int q = col4row[tid];
    int pos = 0;
    for (int i2 = 0; i2 < N_; ++i2) pos += (col4row[i2] < q) ? 1 : 0;
    outR[b * N_ + pos] = (float)q;    // query index
    outC[b * N_ + pos] = (float)tid;  // target index
  }
}

// ---------------- launch ---------------------------------------------------

extern "C" void kernel_launch(void* const* d_in, const int* in_sizes, int n_in,
                              void* d_out, int out_size, void* d_ws, size_t ws_size,
                              hipStream_t stream) {
  (void)in_sizes; (void)n_in; (void)out_size;

  const float* PB = (const float*)d_in[0];   // (B,Q,4)
  const float* PK = (const float*)d_in[1];   // (B,Q,17,2)
  const float* TB = (const float*)d_in[2];   // (B,N,4)
  const float* TK = (const float*)d_in[3];   // (B,N,17,2)

  float* out  = (float*)d_out;
  float* Cout = out;                          // B*Q*N
  float* outR = out + (size_t)B_ * Q_ * N_;   // B*N
  float* outC = outR + (size_t)B_ * N_;       // B*N

  const size_t ctBytes = (size_t)B_ * Q_ * N_ * sizeof(float);
  const bool useCT = (d_ws != nullptr) && (ws_size >= ctBytes);
  float* CTws = useCT ? (float*)d_ws : nullptr;

  hm_cost_kernel<<<dim3(B_ * QBLOCKS), dim3(256), 0, stream>>>(
      PB, PK, TB, TK, Cout, CTws, useCT ? 1 : 0);

  if (useCT)
    hm_hungarian_kernel<true><<<dim3(B_), dim3(512), 0, stream>>>(CTws, outR, outC);
  else
    hm_hungarian_kernel<false><<<dim3(B_), dim3(512), 0, stream>>>(Cout, outR, outC);
}